// MemN2N_57071525429503
// MI455X (gfx1250) — compile-verified
//
#include <hip/hip_runtime.h>
#include <hip/hip_bf16.h>
#include <math.h>

// Problem constants (from reference setup_inputs)
#define B_   128
#define M_   200
#define S_   20
#define D_   128
#define NC_  10000
#define CL_  10

typedef __attribute__((ext_vector_type(2))) float v2f;
typedef __attribute__((ext_vector_type(4))) float v4f;
typedef __attribute__((ext_vector_type(8))) float v8f;

// ---------------------------------------------------------------------------
// u[b,d] = sum_s A[queries[b,s], d], with embedding row 0 forced to zero
// ---------------------------------------------------------------------------
__global__ void memn2n_query_sum(const int* __restrict__ queries,
                                 const float* __restrict__ A_tab,
                                 float* __restrict__ u) {
    const int b = blockIdx.x;      // 0..127
    const int d = threadIdx.x;     // 0..127
    float acc = 0.0f;
    const int* q = queries + b * S_;
    #pragma unroll
    for (int s = 0; s < S_; ++s) {
        const int idx = q[s];
        if (idx != 0) acc += A_tab[(size_t)idx * D_ + d];
    }
    u[b * D_ + d] = acc;
}

// ---------------------------------------------------------------------------
// m[b,mi,d] = sum_s A[stories[b,mi,s], d]   (row 0 zeroed)
// one block per (b,mi) row; 128 threads = d
// ---------------------------------------------------------------------------
__global__ void memn2n_story_sum(const int* __restrict__ stories,
                                 const float* __restrict__ A_tab,
                                 float* __restrict__ m) {
    const int row = blockIdx.x;    // 0 .. B*M-1
    const int d = threadIdx.x;
    float acc = 0.0f;
    const int* st = stories + (size_t)row * S_;
    #pragma unroll
    for (int s = 0; s < S_; ++s) {
        const int idx = st[s];
        if (idx != 0) acc += A_tab[(size_t)idx * D_ + d];
    }
    m[(size_t)row * D_ + d] = acc;
}

// ---------------------------------------------------------------------------
// cand_sum[c,d] = sum_cl W[candidates[c,cl], d]   (row 0 zeroed)
// ---------------------------------------------------------------------------
__global__ void memn2n_cand_sum(const int* __restrict__ candidates,
                                const float* __restrict__ W_tab,
                                float* __restrict__ cand) {
    const int c = blockIdx.x;      // 0..NC-1
    const int d = threadIdx.x;
    float acc = 0.0f;
    const int* cd = candidates + (size_t)c * CL_;
    #pragma unroll
    for (int cl = 0; cl < CL_; ++cl) {
        const int idx = cd[cl];
        if (idx != 0) acc += W_tab[(size_t)idx * D_ + d];
    }
    cand[(size_t)c * D_ + d] = acc;
}

// ---------------------------------------------------------------------------
// 3 memory hops; one block (256 threads = 8 waves) per batch row.
// u kept in LDS; m streamed from L2.
// ---------------------------------------------------------------------------
__global__ void memn2n_hops(const float* __restrict__ m,
                            const float* __restrict__ H_w,
                            float* __restrict__ u_ws) {
    __shared__ float u_s[D_];
    __shared__ float att[M_];
    __shared__ float s_red[2];     // [0]=max, [1]=sum

    const int b    = blockIdx.x;
    const int tid  = threadIdx.x;
    const int lane = tid & 31;
    const int wave = tid >> 5;     // 0..7
    const float* mb = m + (size_t)b * M_ * D_;

    if (tid < D_) u_s[tid] = u_ws[b * D_ + tid];
    __syncthreads();

    for (int hop = 0; hop < 3; ++hop) {
        // ---- dotted[mi] = <m[b,mi,:], u> : one wave per row, strided ----
        for (int mi = wave; mi < M_; mi += 8) {
            const float* row = mb + mi * D_;
            float p = 0.0f;
            #pragma unroll
            for (int k = 0; k < D_ / 32; ++k)
                p += row[lane + 32 * k] * u_s[lane + 32 * k];
            #pragma unroll
            for (int off = 16; off >= 1; off >>= 1)
                p += __shfl_xor(p, off, 32);
            if (lane == 0) att[mi] = p;
        }
        __syncthreads();

        // ---- softmax max (wave 0) ----
        if (wave == 0) {
            float mx = -3.402823466e38f;
            for (int i = lane; i < M_; i += 32) mx = fmaxf(mx, att[i]);
            #pragma unroll
            for (int off = 16; off >= 1; off >>= 1)
                mx = fmaxf(mx, __shfl_xor(mx, off, 32));
            if (lane == 0) s_red[0] = mx;
        }
        __syncthreads();
        const float mx = s_red[0];
        for (int i = tid; i < M_; i += 256)
            att[i] = expf(att[i] - mx);
        __syncthreads();

        // ---- softmax sum (wave 0) ----
        if (wave == 0) {
            float sm = 0.0f;
            for (int i = lane; i < M_; i += 32) sm += att[i];
            #pragma unroll
            for (int off = 16; off >= 1; off >>= 1)
                sm += __shfl_xor(sm, off, 32);
            if (lane == 0) s_red[1] = sm;
        }
        __syncthreads();
        const float inv = 1.0f / s_red[1];

        // ---- o[d] = sum_mi probs*m ; u = tanh(H_w @ u + o) ----
        float u_new = 0.0f;
        if (tid < D_) {
            float o = 0.0f;
            for (int mi = 0; mi < M_; ++mi)
                o += att[mi] * mb[mi * D_ + tid];     // coalesced across d
            o *= inv;
            float h = 0.0f;
            const float* hr = H_w + tid * D_;         // H_w row d (u @ H_w.T)
            #pragma unroll 8
            for (int k = 0; k < D_; ++k)
                h += hr[k] * u_s[k];
            u_new = tanhf(h + o);
        }
        __syncthreads();
        if (tid < D_) u_s[tid] = u_new;
        __syncthreads();
    }

    if (tid < D_) u_ws[b * D_ + tid] = u_s[tid];
}

// ---------------------------------------------------------------------------
// logits[b,c] = sum_d u[b,d] * cand[c,d]  via V_WMMA_F32_16X16X4_F32.
// One wave per 16-wide candidate tile; wave computes all 8 row tiles (M=128).
// Software-pipelined: fragments for k-step kk+1 are loaded BEFORE the 8
// independent WMMAs of step kk, so the backend can clause the 9 loads and
// overlap them with the matrix pipe instead of draining loadcnt per WMMA.
// A frag (ISA 7.12.2, 32-bit A 16x4): lanes 0-15 {K0,K1}, 16-31 {K2,K3}
// B frag (4x16): v0 = {row K0 | row K1}, v1 = {row K2 | row K3}
// C/D (16x16 f32): VGPR j = row j (lanes 0-15) / row 8+j (lanes 16-31)
// ---------------------------------------------------------------------------
__global__ void memn2n_logits_wmma(const float* __restrict__ u,
                                   const float* __restrict__ cand,
                                   float* __restrict__ out) {
    const int lane = threadIdx.x;       // 32 threads = one wave, EXEC all-1s
    const int cb   = blockIdx.x * 16;   // candidate tile base
    const int n    = lane & 15;         // tile column
    const int kh   = lane >> 4;         // half-wave selector (0/1)

    v8f acc[8];
    const v8f zero = {0.f, 0.f, 0.f, 0.f, 0.f, 0.f, 0.f, 0.f};
    #pragma unroll
    for (int rt = 0; rt < 8; ++rt) acc[rt] = zero;

    // B[k, n] = cand[cb+n, k] (cand is [NC, D] row-major = B^T)
    const float* crow  = cand + (size_t)(cb + n) * D_;          // 512B-aligned row
    // A[m, k] = u[row, k]; per-lane element base (8B-aligned)
    const float* abase = u + (size_t)n * D_ + kh * 2;

    v2f aCur[8];
    v2f bCur;
    {   // prologue: fragments for kk = 0
        const v4f q = *(const v4f*)(crow);                      // k = 0..3
        bCur.x = kh ? q.y : q.x;                                // k = kh
        bCur.y = kh ? q.w : q.z;                                // k = 2 + kh
        #pragma unroll
        for (int rt = 0; rt < 8; ++rt)
            aCur[rt] = *(const v2f*)(abase + rt * 16 * D_);     // k = 2*kh, 2*kh+1
    }

    #pragma unroll
    for (int kk = 0; kk < D_ / 4; ++kk) {
        v2f aNxt[8];
        v2f bNxt;
        if (kk < D_ / 4 - 1) {          // issue next step's loads first
            const int kb = (kk + 1) * 4;
            const v4f q = *(const v4f*)(crow + kb);
            bNxt.x = kh ? q.y : q.x;
            bNxt.y = kh ? q.w : q.z;
            #pragma unroll
            for (int rt = 0; rt < 8; ++rt)
                aNxt[rt] = *(const v2f*)(abase + rt * 16 * D_ + kb);
        }
        #pragma unroll
        for (int rt = 0; rt < 8; ++rt)
            acc[rt] = __builtin_amdgcn_wmma_f32_16x16x4_f32(
                false, aCur[rt], false, bCur, (short)0, acc[rt], false, false);
        if (kk < D_ / 4 - 1) {
            bCur = bNxt;
            #pragma unroll
            for (int rt = 0; rt < 8; ++rt) aCur[rt] = aNxt[rt];
        }
    }

    // store D tiles
    const int col = cb + n;
    #pragma unroll
    for (int rt = 0; rt < 8; ++rt) {
        #pragma unroll
        for (int j = 0; j < 8; ++j) {
            const int row = rt * 16 + kh * 8 + j;
            out[(size_t)row * NC_ + col] = acc[rt][j];
        }
    }
}

// ---------------------------------------------------------------------------
extern "C" void kernel_launch(void* const* d_in, const int* in_sizes, int n_in,
                              void* d_out, int out_size, void* d_ws, size_t ws_size,
                              hipStream_t stream) {
    (void)in_sizes; (void)n_in; (void)out_size; (void)ws_size;

    const int*   stories    = (const int*)d_in[0];   // [B, M, S]
    const int*   queries    = (const int*)d_in[1];   // [B, S]
    const int*   candidates = (const int*)d_in[2];   // [NC, CL]
    const float* A_tab      = (const float*)d_in[3]; // [V, D]
    const float* W_tab      = (const float*)d_in[4]; // [V, D]
    const float* H_w        = (const float*)d_in[5]; // [D, D]
    float*       out        = (float*)d_out;         // [B, NC]

    float* ws   = (float*)d_ws;
    float* u    = ws;                                  // B*D       = 16384
    float* m    = u + B_ * D_;                         // B*M*D     = 3,276,800
    float* cand = m + (size_t)B_ * M_ * D_;            // NC*D      = 1,280,000

    memn2n_query_sum<<<B_,        D_, 0, stream>>>(queries, A_tab, u);
    memn2n_story_sum<<<B_ * M_,   D_, 0, stream>>>(stories, A_tab, m);
    memn2n_cand_sum <<<NC_,       D_, 0, stream>>>(candidates, W_tab, cand);
    memn2n_hops     <<<B_,       256, 0, stream>>>(m, H_w, u);
    memn2n_logits_wmma<<<NC_ / 16, 32, 0, stream>>>(u, cand, out);
}